// ExpDock_68650757259711
// MI455X (gfx1250) — compile-verified
//
#include <hip/hip_runtime.h>
#include <hip/hip_bf16.h>

typedef float v2f __attribute__((ext_vector_type(2)));
typedef float v8f __attribute__((ext_vector_type(8)));

#define NROWS 16384
#define DDIM  128
#define KHEAD 10
#define KPAD  16
#define NB    16
#define SCALE 0.08838834764831845f   // 1/sqrt(128)

// ---------------- Kernel 1: column-sum partials over 256-row chunks ----------
__global__ __launch_bounds__(128) void k1_colsum(const float* __restrict__ H1,
                                                 const float* __restrict__ H2,
                                                 float* __restrict__ colsum) {
  const int chunk = blockIdx.x;   // 0..63
  const int m     = blockIdx.y;   // 0=H1, 1=H2
  const int b     = blockIdx.z;   // 0..15
  const int t     = threadIdx.x;  // 0..127 (column)
  const float* H = m ? H2 : H1;
  const float* p = H + ((long long)b * NROWS + (long long)chunk * 256) * DDIM + t;
  float acc = 0.f;
  for (int n = 0; n < 256; ++n) acc += p[(long long)n * DDIM];
  colsum[(((long long)(m * NB + b)) * 64 + chunk) * DDIM + t] = acc;
}

// ---------------- Kernel 2: reduce 64 partials -> mean -----------------------
__global__ __launch_bounds__(128) void k2_mean(const float* __restrict__ colsum,
                                               float* __restrict__ hbar) {
  const int m = blockIdx.x, b = blockIdx.y, t = threadIdx.x;
  const float* p = colsum + ((long long)(m * NB + b)) * 64 * DDIM + t;
  float acc = 0.f;
  for (int ch = 0; ch < 64; ++ch) acc += p[ch * DDIM];
  hbar[(m * NB + b) * DDIM + t] = acc * (1.0f / (float)NROWS);
}

// ---------------- Kernel 3: v[side,b,k,:] = W_side[k] @ hbar_other * scale ---
__global__ __launch_bounds__(128) void k3_v(const float* __restrict__ W1,
                                            const float* __restrict__ W2,
                                            const float* __restrict__ hbar,
                                            float* __restrict__ vpad) {
  const int k = blockIdx.x;   // 0..15 (padded head)
  const int s = blockIdx.y;   // side
  const int b = blockIdx.z;
  const int t = threadIdx.x;  // 0..127 (output dim)
  __shared__ float hb[DDIM];
  hb[t] = hbar[((1 - s) * NB + b) * DDIM + t];  // side0 uses mean(H2)
  __syncthreads();
  float acc = 0.f;
  if (k < KHEAD) {
    const float* W = s ? W2 : W1;
    const float* wr = W + ((long long)k * DDIM + t) * DDIM;
    #pragma unroll 8
    for (int e = 0; e < DDIM; ++e) acc += wr[e] * hb[e];
  }
  vpad[(((long long)(s * NB + b)) * KPAD + k) * DDIM + t] = acc * SCALE;
}

// ---------------- Kernel 4: WMMA scores + online softmax partials ------------
// grid (256 tiles, 2 sides, 16 b), 128 threads (4 waves of 32)
__global__ __launch_bounds__(128) void k4_score(const float* __restrict__ H1,
                                                const float* __restrict__ H2,
                                                const float* __restrict__ X1,
                                                const float* __restrict__ X2,
                                                const float* __restrict__ vpad,
                                                float* __restrict__ bpart) {
  constexpr int ROWS = 64, PAD = 132;
  __shared__ float lh[ROWS * PAD];    // 33792 B
  __shared__ float lv[KPAD * PAD];    //  8448 B
  __shared__ float red[4][KPAD][5];   //  1280 B

  const int tile = blockIdx.x;
  const int side = blockIdx.y;
  const int b    = blockIdx.z;
  const int tid  = threadIdx.x;
  const int wave = tid >> 5;
  const int lane = tid & 31;
  const int hi   = lane >> 4;   // half-wave select
  const int lo   = lane & 15;

  const float* H = side ? H2 : H1;
  const float* X = side ? X2 : X1;
  const long long nbase = (long long)tile * ROWS;
  const float* Hb = H + ((long long)b * NROWS + nbase) * DDIM;
  const float* Vb = vpad + ((long long)(side * NB + b)) * KPAD * DDIM;

  // stage H tile (64x128) coalesced into padded LDS
  #pragma unroll
  for (int it = 0; it < 16; ++it) {
    int flat4 = it * 128 + tid;           // 0..2047 float4s
    int row = flat4 >> 5;
    int c4  = (flat4 & 31) << 2;
    float4 v = *(const float4*)(Hb + (long long)row * DDIM + c4);
    *(float4*)&lh[row * PAD + c4] = v;
  }
  // stage v (16x128)
  #pragma unroll
  for (int it = 0; it < 4; ++it) {
    int flat4 = it * 128 + tid;           // 0..511
    int row = flat4 >> 5;
    int c4  = (flat4 & 31) << 2;
    float4 v = *(const float4*)(Vb + row * DDIM + c4);
    *(float4*)&lv[row * PAD + c4] = v;
  }
  __syncthreads();

  // 16(n) x 16(head) score tile per wave via 32 chained f32 WMMAs over d=128
  const float* arow = &lh[(wave * 16 + lo) * PAD];  // A: M = lane&15
  const float* brow = &lv[lo * PAD];                // B: N(head) = lane&15
  v8f acc = {};
  #pragma unroll
  for (int k0 = 0; k0 < DDIM; k0 += 4) {
    v2f a  = *(const v2f*)(arow + k0 + 2 * hi);
    v2f bv = *(const v2f*)(brow + k0 + 2 * hi);
    acc = __builtin_amdgcn_wmma_f32_16x16x4_f32(false, a, false, bv,
                                                (short)0, acc, false, false);
  }

  // online softmax partial: lane covers head=lo, rows wave*16 + 8*hi + j
  float m = acc[0];
  #pragma unroll
  for (int j = 1; j < 8; ++j) m = fmaxf(m, acc[j]);
  m = fmaxf(m, __shfl_xor(m, 16, 32));
  float s = 0.f, w0 = 0.f, w1 = 0.f, w2 = 0.f;
  const float* Xr = X + ((long long)b * NROWS + nbase + wave * 16 + 8 * hi) * 3;
  #pragma unroll
  for (int j = 0; j < 8; ++j) {
    float e = __expf(acc[j] - m);
    s  += e;
    w0 += e * Xr[j * 3 + 0];
    w1 += e * Xr[j * 3 + 1];
    w2 += e * Xr[j * 3 + 2];
  }
  s  += __shfl_xor(s, 16, 32);
  w0 += __shfl_xor(w0, 16, 32);
  w1 += __shfl_xor(w1, 16, 32);
  w2 += __shfl_xor(w2, 16, 32);
  if (hi == 0) {
    red[wave][lo][0] = m;  red[wave][lo][1] = s;
    red[wave][lo][2] = w0; red[wave][lo][3] = w1; red[wave][lo][4] = w2;
  }
  __syncthreads();
  if (tid < KPAD) {  // merge 4 wave partials per head (deterministic order)
    float M = red[0][tid][0], S = red[0][tid][1];
    float W0 = red[0][tid][2], W1 = red[0][tid][3], W2 = red[0][tid][4];
    #pragma unroll
    for (int w = 1; w < 4; ++w) {
      float m2 = red[w][tid][0];
      float nM = fmaxf(M, m2);
      float f1 = __expf(M - nM), f2 = __expf(m2 - nM);
      S  = S  * f1 + red[w][tid][1] * f2;
      W0 = W0 * f1 + red[w][tid][2] * f2;
      W1 = W1 * f1 + red[w][tid][3] * f2;
      W2 = W2 * f1 + red[w][tid][4] * f2;
      M = nM;
    }
    float* o = bpart + ((((long long)(side * NB + b)) * 256 + tile) * KPAD + tid) * 5;
    o[0] = M; o[1] = S; o[2] = W0; o[3] = W1; o[4] = W2;
  }
}

// ---------------- Kernel 5: fold partials -> Y1,Y2; Kabsch; write output -----
__global__ __launch_bounds__(64) void k5_final(const float* __restrict__ bpart,
                                               float* __restrict__ out) {
  const int b = blockIdx.x;
  const int tid = threadIdx.x;
  __shared__ float Y[2][KPAD][3];
  if (tid < 32) {
    const int side = tid >> 4, head = tid & 15;
    const float* base = bpart + (((long long)(side * NB + b)) * 256) * KPAD * 5 + head * 5;
    float M = -3.402823466e38f, S = 0.f, W0 = 0.f, W1 = 0.f, W2 = 0.f;
    for (int t = 0; t < 256; ++t) {
      const float* p = base + (long long)t * KPAD * 5;
      float m2 = p[0];
      float nM = fmaxf(M, m2);
      float f1 = __expf(M - nM), f2 = __expf(m2 - nM);
      S  = S  * f1 + p[1] * f2;
      W0 = W0 * f1 + p[2] * f2;
      W1 = W1 * f1 + p[3] * f2;
      W2 = W2 * f1 + p[4] * f2;
      M = nM;
    }
    float inv = 1.0f / S;
    Y[side][head][0] = W0 * inv;
    Y[side][head][1] = W1 * inv;
    Y[side][head][2] = W2 * inv;
  }
  __syncthreads();
  if (tid != 0) return;

  // Kabsch: align P=Y1 onto Q=Y2 (row-vector convention, matches reference)
  float P[KHEAD][3], Q[KHEAD][3], c1[3] = {0,0,0}, c2[3] = {0,0,0};
  for (int k = 0; k < KHEAD; ++k)
    for (int c = 0; c < 3; ++c) {
      P[k][c] = Y[0][k][c]; Q[k][c] = Y[1][k][c];
      c1[c] += P[k][c]; c2[c] += Q[k][c];
    }
  for (int c = 0; c < 3; ++c) { c1[c] *= 0.1f; c2[c] *= 0.1f; }

  float Hm[3][3] = {{0,0,0},{0,0,0},{0,0,0}};
  for (int k = 0; k < KHEAD; ++k)
    for (int i = 0; i < 3; ++i)
      for (int j = 0; j < 3; ++j)
        Hm[i][j] += (P[k][i] - c1[i]) * (Q[k][j] - c2[j]);

  // A = Hm^T Hm, Jacobi eigendecomposition -> V, lambda
  float A[3][3], V[3][3] = {{1,0,0},{0,1,0},{0,0,1}};
  for (int i = 0; i < 3; ++i)
    for (int j = 0; j < 3; ++j)
      A[i][j] = Hm[0][i]*Hm[0][j] + Hm[1][i]*Hm[1][j] + Hm[2][i]*Hm[2][j];
  for (int sweep = 0; sweep < 12; ++sweep) {
    for (int pair = 0; pair < 3; ++pair) {
      const int p = (pair == 2) ? 1 : 0;
      const int q = (pair == 0) ? 1 : 2;
      float apq = A[p][q];
      if (fabsf(apq) > 1e-18f) {
        float tau = (A[q][q] - A[p][p]) / (2.0f * apq);
        float tt = (tau >= 0.f ? 1.f : -1.f) / (fabsf(tau) + sqrtf(1.f + tau*tau));
        float cr = 1.0f / sqrtf(1.f + tt*tt);
        float sr = tt * cr;
        for (int mm = 0; mm < 3; ++mm) {
          float amp = A[mm][p], amq = A[mm][q];
          A[mm][p] = cr*amp - sr*amq; A[mm][q] = sr*amp + cr*amq;
        }
        for (int mm = 0; mm < 3; ++mm) {
          float apm = A[p][mm], aqm = A[q][mm];
          A[p][mm] = cr*apm - sr*aqm; A[q][mm] = sr*apm + cr*aqm;
        }
        for (int mm = 0; mm < 3; ++mm) {
          float vmp = V[mm][p], vmq = V[mm][q];
          V[mm][p] = cr*vmp - sr*vmq; V[mm][q] = sr*vmp + cr*vmq;
        }
      }
    }
  }
  // sort eigenpairs descending
  float lam[3] = {A[0][0], A[1][1], A[2][2]};
  for (int i = 0; i < 2; ++i)
    for (int j = 0; j < 2 - i; ++j)
      if (lam[j] < lam[j+1]) {
        float tl = lam[j]; lam[j] = lam[j+1]; lam[j+1] = tl;
        for (int mm = 0; mm < 3; ++mm) {
          float tv = V[mm][j]; V[mm][j] = V[mm][j+1]; V[mm][j+1] = tv;
        }
      }
  // U columns: u_i = Hm*v_i / |Hm*v_i|; u2 = u0 x u1 (det U = +1)
  float U[3][3];
  for (int col = 0; col < 2; ++col) {
    float x[3];
    for (int i = 0; i < 3; ++i)
      x[i] = Hm[i][0]*V[0][col] + Hm[i][1]*V[1][col] + Hm[i][2]*V[2][col];
    float nrm = sqrtf(x[0]*x[0] + x[1]*x[1] + x[2]*x[2]) + 1e-30f;
    for (int i = 0; i < 3; ++i) U[i][col] = x[i] / nrm;
  }
  {  // re-orthogonalize u1 against u0
    float dp = U[0][0]*U[0][1] + U[1][0]*U[1][1] + U[2][0]*U[2][1];
    float x[3];
    for (int i = 0; i < 3; ++i) x[i] = U[i][1] - dp * U[i][0];
    float nrm = sqrtf(x[0]*x[0] + x[1]*x[1] + x[2]*x[2]) + 1e-30f;
    for (int i = 0; i < 3; ++i) U[i][1] = x[i] / nrm;
  }
  U[0][2] = U[1][0]*U[2][1] - U[2][0]*U[1][1];
  U[1][2] = U[2][0]*U[0][1] - U[0][0]*U[2][1];
  U[2][2] = U[0][0]*U[1][1] - U[1][0]*U[0][1];

  float detV = V[0][0]*(V[1][1]*V[2][2] - V[1][2]*V[2][1])
             - V[0][1]*(V[1][0]*V[2][2] - V[1][2]*V[2][0])
             + V[0][2]*(V[1][0]*V[2][1] - V[1][1]*V[2][0]);
  float sgn = (detV >= 0.f) ? 1.f : -1.f;   // det(U)=+1 by construction

  float R[3][3];
  for (int i = 0; i < 3; ++i)
    for (int j = 0; j < 3; ++j)
      R[i][j] = U[i][0]*V[j][0] + U[i][1]*V[j][1] + sgn*U[i][2]*V[j][2];

  float tvec[3];
  for (int c = 0; c < 3; ++c)
    tvec[c] = c2[c] - (c1[0]*R[0][c] + c1[1]*R[1][c] + c1[2]*R[2][c]);

  float* ob = out + (long long)b * 3 * KHEAD * 3;
  for (int k = 0; k < KHEAD; ++k)
    for (int c = 0; c < 3; ++c) {
      float ya = P[k][0]*R[0][c] + P[k][1]*R[1][c] + P[k][2]*R[2][c] + tvec[c];
      ob[(0 * KHEAD + k) * 3 + c] = P[k][c];
      ob[(1 * KHEAD + k) * 3 + c] = Q[k][c];
      ob[(2 * KHEAD + k) * 3 + c] = ya;
    }
}

extern "C" void kernel_launch(void* const* d_in, const int* in_sizes, int n_in,
                              void* d_out, int out_size, void* d_ws, size_t ws_size,
                              hipStream_t stream) {
  const float* H1 = (const float*)d_in[0];
  const float* H2 = (const float*)d_in[1];
  const float* X1 = (const float*)d_in[2];
  const float* X2 = (const float*)d_in[3];
  const float* W1 = (const float*)d_in[4];
  const float* W2 = (const float*)d_in[5];
  float* ws = (float*)d_ws;

  float* colsum = ws;                          // 2*16*64*128   = 262144 f
  float* hbar   = colsum + 2*16*64*128;        // 2*16*128      = 4096 f
  float* vpad   = hbar   + 2*16*128;           // 2*16*16*128   = 65536 f
  float* bpart  = vpad   + 2*16*16*128;        // 2*16*256*16*5 = 655360 f

  k1_colsum<<<dim3(64, 2, 16), 128, 0, stream>>>(H1, H2, colsum);
  k2_mean  <<<dim3(2, 16),     128, 0, stream>>>(colsum, hbar);
  k3_v     <<<dim3(16, 2, 16), 128, 0, stream>>>(W1, W2, hbar, vpad);
  k4_score <<<dim3(256, 2, 16),128, 0, stream>>>(H1, H2, X1, X2, vpad, bpart);
  k5_final <<<dim3(16),        64,  0, stream>>>(bpart, (float*)d_out);
}